// Dynamic_emsemble_15049565405701
// MI455X (gfx1250) — compile-verified
//
#include <hip/hip_runtime.h>

typedef __attribute__((ext_vector_type(2))) float v2f;
typedef __attribute__((ext_vector_type(8))) float v8f;

#define B_   4
#define C_   68
#define F0   4      // first used feature channel
#define PT   2048
#define PS   8192
#define KSEL 4
#define FLT_BIG 3.4e38f

// ---------------------------------------------------------------------------
// Kernel 1: sq[b][q] = sum_f Source[b, 4+f, q]^2   (f = 0..63)
// ---------------------------------------------------------------------------
__global__ void sqnorm_kernel(const float* __restrict__ S, float* __restrict__ sq) {
  int q = blockIdx.x * blockDim.x + threadIdx.x;
  int b = blockIdx.y;
  if (q >= PS) return;
  const float* p = S + ((size_t)b * C_ + F0) * PS + q;
  float acc = 0.f;
  #pragma unroll 8
  for (int f = 0; f < 64; ++f) {
    float v = p[(size_t)f * PS];
    acc = fmaf(v, v, acc);
  }
  sq[b * PS + q] = acc;
}

// ---------------------------------------------------------------------------
// Kernel 2: per (b, 16-row tile of target points) find top-4 nearest source
// indices using f32 WMMA for the inner-product matrix.
//
// WMMA f32 16x16x4 layouts (ISA 7.12.2, 32-bit matrices, wave32):
//   A (16x4, MxK):  lane L: M = L%16 ; VGPR v, half h=L/16 -> K = v + 2h
//   B (4x16, KxN):  lane L: N = L%16 ; VGPR v, half h      -> K = v + 2h
//   C/D (16x16):    lane L: N = L%16 ; VGPR v, half h      -> M = v + 8h
// ---------------------------------------------------------------------------
__launch_bounds__(128)
__global__ void knn_kernel(const float* __restrict__ T, const float* __restrict__ S,
                           const float* __restrict__ sq, int* __restrict__ out_idx) {
  const int b    = blockIdx.y;
  const int p0   = blockIdx.x * 16;
  const int lane = threadIdx.x & 31;
  const int wave = threadIdx.x >> 5;
  const int half = lane >> 4;   // 0/1: selects K pair (A,B) / M group (C,D)
  const int ln   = lane & 15;   // M for A ; N for B,C,D

  // Preload full 16x64 A tile: 16 K-steps of 4, a[s] covers K = 4s+2h+{0,1}.
  v2f a[16];
  #pragma unroll
  for (int s = 0; s < 16; ++s) {
    int f = F0 + 4 * s + 2 * half;
    const float* tp = T + ((size_t)b * C_ + f) * PT + p0 + ln;
    a[s].x = tp[0];
    a[s].y = tp[PT];   // next feature row
  }

  // Per-lane running top-4 for each of the 8 C-rows this lane holds.
  float best[8][KSEL];
  int   bidx[8][KSEL];
  #pragma unroll
  for (int v = 0; v < 8; ++v)
    #pragma unroll
    for (int j = 0; j < KSEL; ++j) { best[v][j] = FLT_BIG; bidx[v][j] = 0; }

  const int qbeg = wave * (PS / 4);
  const int qend = qbeg + (PS / 4);
  for (int q0 = qbeg; q0 < qend; q0 += 16) {
    // Load B tile fragments (64x16 slab of sf).
    v2f bb[16];
    #pragma unroll
    for (int s = 0; s < 16; ++s) {
      int f = F0 + 4 * s + 2 * half;
      const float* sp = S + ((size_t)b * C_ + f) * PS + q0 + ln;
      bb[s].x = sp[0];
      bb[s].y = sp[PS];
    }

    // 16 chained f32 WMMAs: acc = A(16x64) * B(64x16)
    v8f acc = {};
    #pragma unroll
    for (int s = 0; s < 16; ++s) {
      acc = __builtin_amdgcn_wmma_f32_16x16x4_f32(
          /*neg_a=*/false, a[s], /*neg_b=*/false, bb[s],
          /*c_mod=*/(short)0, acc, /*reuse_a=*/false, /*reuse_b=*/false);
    }

    const float sqv  = sq[b * PS + q0 + ln];  // same column for both halves
    const int   qidx = q0 + ln;

    #pragma unroll
    for (int v = 0; v < 8; ++v) {
      float sc = fmaf(-2.f, acc[v], sqv);     // ||s||^2 - 2 t.s  (row-const dropped)
      if (sc < best[v][3]) {
        if (sc < best[v][2]) {
          best[v][3] = best[v][2]; bidx[v][3] = bidx[v][2];
          if (sc < best[v][1]) {
            best[v][2] = best[v][1]; bidx[v][2] = bidx[v][1];
            if (sc < best[v][0]) {
              best[v][1] = best[v][0]; bidx[v][1] = bidx[v][0];
              best[v][0] = sc; bidx[v][0] = qidx;
            } else { best[v][1] = sc; bidx[v][1] = qidx; }
          } else { best[v][2] = sc; bidx[v][2] = qidx; }
        } else { best[v][3] = sc; bidx[v][3] = qidx; }
      }
    }
  }

  // Merge candidates across lanes and waves through LDS.
  __shared__ float ls[4][16][16][KSEL];
  __shared__ int   li[4][16][16][KSEL];
  #pragma unroll
  for (int v = 0; v < 8; ++v) {
    int row = v + 8 * half;
    #pragma unroll
    for (int j = 0; j < KSEL; ++j) {
      ls[wave][row][ln][j] = best[v][j];
      li[wave][row][ln][j] = bidx[v][j];
    }
  }
  __syncthreads();

  if (threadIdx.x < 16) {
    const int row = threadIdx.x;
    float rs[KSEL]; int ri[KSEL];
    #pragma unroll
    for (int j = 0; j < KSEL; ++j) { rs[j] = FLT_BIG; ri[j] = 0; }
    for (int w = 0; w < 4; ++w)
      for (int n = 0; n < 16; ++n)
        #pragma unroll
        for (int j = 0; j < KSEL; ++j) {
          float sc = ls[w][row][n][j];
          int   id = li[w][row][n][j];
          if (sc < rs[3]) {
            if (sc < rs[2]) {
              rs[3] = rs[2]; ri[3] = ri[2];
              if (sc < rs[1]) {
                rs[2] = rs[1]; ri[2] = ri[1];
                if (sc < rs[0]) {
                  rs[1] = rs[0]; ri[1] = ri[0]; rs[0] = sc; ri[0] = id;
                } else { rs[1] = sc; ri[1] = id; }
              } else { rs[2] = sc; ri[2] = id; }
            } else { rs[3] = sc; ri[3] = id; }
          }
        }
    int* o = out_idx + ((size_t)b * PT + p0 + row) * KSEL;
    o[0] = ri[0]; o[1] = ri[1]; o[2] = ri[2]; o[3] = ri[3];
  }
}

// ---------------------------------------------------------------------------
// Kernel 3: out[b,c,p,k] = Source[b,c, idx[b,p,k]]
// ---------------------------------------------------------------------------
__global__ void gather_kernel(const float* __restrict__ S, const int* __restrict__ idx,
                              float* __restrict__ out) {
  size_t t = (size_t)blockIdx.x * blockDim.x + threadIdx.x;   // over B*C*PT
  if (t >= (size_t)B_ * C_ * PT) return;
  int p  = (int)(t % PT);
  int bc = (int)(t / PT);        // b*C + c
  int b  = bc / C_;
  const int4 id = *(const int4*)(idx + ((size_t)b * PT + p) * KSEL);
  const float* src = S + (size_t)bc * PS;
  float4 r;
  r.x = src[id.x]; r.y = src[id.y]; r.z = src[id.z]; r.w = src[id.w];
  *(float4*)(out + t * KSEL) = r;
}

extern "C" void kernel_launch(void* const* d_in, const int* in_sizes, int n_in,
                              void* d_out, int out_size, void* d_ws, size_t ws_size,
                              hipStream_t stream) {
  const float* T = (const float*)d_in[0];   // Target (4,68,2048) f32
  const float* S = (const float*)d_in[1];   // Source (4,68,8192) f32
  // d_in[2] = s_num (==4), compile-time constant KSEL

  float* sq  = (float*)d_ws;                                        // 4*8192 f32
  int*   idx = (int*)((char*)d_ws + (size_t)B_ * PS * sizeof(float)); // 4*2048*4 i32
  float* out = (float*)d_out;

  sqnorm_kernel<<<dim3(PS / 256, B_), 256, 0, stream>>>(S, sq);
  knn_kernel<<<dim3(PT / 16, B_), 128, 0, stream>>>(T, S, sq, idx);
  size_t n = (size_t)B_ * C_ * PT;
  gather_kernel<<<dim3((unsigned)((n + 255) / 256)), 256, 0, stream>>>(S, idx, out);
}